// GatbertSelfAttention_38362647888358
// MI455X (gfx1250) — compile-verified
//
#include <hip/hip_runtime.h>

typedef __attribute__((ext_vector_type(2))) float v2f;
typedef __attribute__((ext_vector_type(4))) float v4f;
typedef __attribute__((ext_vector_type(8))) float v8f;

#define Bc   8
#define Nc   1024
#define Hc   768
#define NHc  12
#define HDc  64
#define Ec   131072
#define MROWS (Bc * Nc)      /* 8192 rows of X/Q/K/V */
#define NSEG  (Bc * Nc)      /* 8192 segments (batch, head-node) */
#define KC    256            /* GEMM k-chunk staged in LDS */

// ======================= kernel 1: zero segment counts ===================
__global__ __launch_bounds__(256)
void gat_zero(int* __restrict__ cnt, int n) {
    int t = blockIdx.x * blockDim.x + threadIdx.x;
    if (t < n) cnt[t] = 0;
}

// ======================= kernel 2: fused QKV GEMM (WMMA f32) =============
// X:(8192,768), W*:(768,768) row-major. Block = 8 waves; all waves share one
// 16-column slab of Wq/Wk/Wv, staged per k-chunk into LDS transposed
// (col-major) so each lane's B fragment is one ds_load_b64.
// V_WMMA_F32_16X16X4_F32 layouts:
//   A 16x4 : lane%16 = row; VGPR0/1 = K{0,1} (lanes 0-15), K{2,3} (16-31).
//   B 4x16 : lane%16 = col; VGPR0/1 = K{0,1} (lanes 0-15), K{2,3} (16-31).
//   C/D    : VGPR r -> row r + 8*(lane/16), col = lane%16.
__global__ __launch_bounds__(256)
void gat_qkv_gemm(const float* __restrict__ X,
                  const float* __restrict__ Wq,
                  const float* __restrict__ Wk,
                  const float* __restrict__ Wv,
                  float* __restrict__ Q,
                  float* __restrict__ K,
                  float* __restrict__ V) {
    __shared__ float wlds[3][16][KC + 2];        // +2 pad: bank-conflict-free, keeps 8B align

    const int tid  = threadIdx.x;
    const int lane = tid & 31;
    const int wave = tid >> 5;
    const int nt   = blockIdx.x;                 // 0..47  (768/16)
    const int mt   = blockIdx.y * 8 + wave;      // 0..511 (8192/16)
    const int half = lane >> 4;                  // 0 or 1
    const int l16  = lane & 15;
    const int row  = mt * 16 + l16;
    const int col  = nt * 16 + l16;

    const float* __restrict__ W[3] = { Wq, Wk, Wv };

    v8f cq = {}; v8f ck = {}; v8f cv = {};
    const float* __restrict__ xrow = X + (size_t)row * Hc;

    for (int kc = 0; kc < Hc; kc += KC) {
        __syncthreads();                         // previous chunk fully consumed
        // ---- stage 3 x (KC x 16) W slabs into LDS, transposed ----
        // elements per chunk = 3*KC*16 = 12288 ; 256 threads -> 48 each
        for (int idx = tid; idx < 3 * KC * 16; idx += 256) {
            const int m  = idx >> 12;            // /4096 = /(KC*16)
            const int r  = idx & 4095;
            const int k  = r >> 4;               // 0..KC-1
            const int c  = r & 15;               // 0..15
            wlds[m][c][k] = W[m][(size_t)(kc + k) * Hc + (nt * 16 + c)];
        }
        __syncthreads();

        // ---- WMMA over this k-chunk ----
        for (int k4 = 0; k4 < KC; k4 += 4) {
            const int ks = k4 + half * 2;
            v2f a  = *(const v2f*)(xrow + kc + ks);
            v2f bq = *(const v2f*)&wlds[0][l16][ks];
            v2f bk = *(const v2f*)&wlds[1][l16][ks];
            v2f bv = *(const v2f*)&wlds[2][l16][ks];
            cq = __builtin_amdgcn_wmma_f32_16x16x4_f32(false, a, false, bq, (short)0, cq, false, false);
            ck = __builtin_amdgcn_wmma_f32_16x16x4_f32(false, a, false, bk, (short)0, ck, false, false);
            cv = __builtin_amdgcn_wmma_f32_16x16x4_f32(false, a, false, bv, (short)0, cv, false, false);
        }
    }

#pragma unroll
    for (int r = 0; r < 8; ++r) {
        const int m = mt * 16 + r + 8 * half;
        Q[(size_t)m * Hc + col] = cq[r];
        K[(size_t)m * Hc + col] = ck[r];
        V[(size_t)m * Hc + col] = cv[r];
    }
}

// ======================= kernel 3: segment histogram =====================
__global__ __launch_bounds__(256)
void gat_count(const int* __restrict__ eidx, int* __restrict__ cnt) {
    int e = blockIdx.x * blockDim.x + threadIdx.x;
    if (e >= Ec) return;
    const int seg = eidx[e] * Nc + eidx[Ec + e];     // b*N + i
    atomicAdd(&cnt[seg], 1);
}

// ======================= kernel 4: exclusive scan (1 block) ==============
__global__ __launch_bounds__(1024)
void gat_scan(const int* __restrict__ cnt, int* __restrict__ base,
              int* __restrict__ cursor) {
    __shared__ int lds[1024];
    const int t = threadIdx.x;                       // NSEG/1024 = 8 per thread
    int local[8];
    int s = 0;
#pragma unroll
    for (int r = 0; r < 8; ++r) { local[r] = cnt[t * 8 + r]; s += local[r]; }
    lds[t] = s;
    __syncthreads();
    for (int off = 1; off < 1024; off <<= 1) {       // Hillis-Steele inclusive
        int v = lds[t];
        int add = (t >= off) ? lds[t - off] : 0;
        __syncthreads();
        lds[t] = v + add;
        __syncthreads();
    }
    int run = (t == 0) ? 0 : lds[t - 1];
#pragma unroll
    for (int r = 0; r < 8; ++r) {
        base[t * 8 + r] = run;
        cursor[t * 8 + r] = run;
        run += local[r];
    }
}

// ======================= kernel 5: scatter edge ids into CSR =============
__global__ __launch_bounds__(256)
void gat_scatter(const int* __restrict__ eidx, int* __restrict__ cursor,
                 int* __restrict__ elist) {
    int e = blockIdx.x * blockDim.x + threadIdx.x;
    if (e >= Ec) return;
    const int seg = eidx[e] * Nc + eidx[Ec + e];
    const int pos = atomicAdd(&cursor[seg], 1);
    elist[pos] = e;
}

// ======================= kernel 6: per-segment fused attention ===========
// One 128-thread block per segment. Online-softmax over edge chunks of 64,
// so any degree is handled. q row == seg row of Q (gathered once into LDS).
// Output written once with plain stores -> no float atomics anywhere.
#define SEGT 128
#define CH   64
__global__ __launch_bounds__(SEGT)
void gat_segment_attn(const float* __restrict__ Q, const float* __restrict__ Km,
                      const float* __restrict__ V, const int* __restrict__ eidx,
                      const int* __restrict__ base, const int* __restrict__ cnt,
                      int* __restrict__ elist, float* __restrict__ out) {
    __shared__ float qs[Hc];                 // this segment's q row (3 KB)
    __shared__ int   rowj[CH];               // tail-node row index per edge
    __shared__ float sc[CH][NHc];            // scores -> weights (3 KB)
    __shared__ float mh[NHc], dh[NHc], fh[NHc];

    const int seg = blockIdx.x;              // b = seg/Nc, i = seg%Nc
    const int t   = threadIdx.x;
    const int beg = base[seg];
    const int end = beg + cnt[seg];
    const int bN  = (seg >> 10) * Nc;        // b * Nc   (Nc == 1024)

    for (int o = t; o < Hc; o += SEGT) qs[o] = Q[(size_t)seg * Hc + o];
    if (t < NHc) { mh[t] = -INFINITY; dh[t] = 0.0f; }

    float acc[Hc / SEGT] = {};               // 6 output accumulators / thread

    for (int c0 = beg; c0 < end; c0 += CH) {
        const int n = min(CH, end - c0);
        __syncthreads();                     // qs ready / prev chunk consumed
        if (t < n) {
            const int e = elist[c0 + t];
            rowj[t] = bN + eidx[2 * Ec + e]; // b*N + j
        }
        __syncthreads();

        // ---- scores for (edge, head) pairs ----
        for (int p = t; p < n * NHc; p += SEGT) {
            const int ec = p / NHc;
            const int h  = p % NHc;
            const float* __restrict__ kb = Km + (size_t)rowj[ec] * Hc + h * HDc;
            const float* __restrict__ qb = &qs[h * HDc];
            float s = 0.0f;
#pragma unroll
            for (int d = 0; d < HDc; d += 4) {
                v4f ka = *(const v4f*)(kb + d);
                s += qb[d] * ka.x + qb[d + 1] * ka.y + qb[d + 2] * ka.z + qb[d + 3] * ka.w;
            }
            sc[ec][h] = s * 0.125f;          // 1/sqrt(64)
        }
        __syncthreads();

        // ---- per-head online max / denom update (threads 0..11) ----
        if (t < NHc) {
            const int h = t;
            float mn = mh[h];
            for (int ec = 0; ec < n; ++ec) mn = fmaxf(mn, sc[ec][h]);
            const float f = __expf(mh[h] - mn);      // 0 when mh == -inf
            float dn = dh[h] * f;
            for (int ec = 0; ec < n; ++ec) {
                const float w = __expf(sc[ec][h] - mn);
                sc[ec][h] = w;
                dn += w;
            }
            mh[h] = mn; dh[h] = dn; fh[h] = f;
        }
        __syncthreads();

        // ---- rescale accumulators and add weighted V rows ----
#pragma unroll
        for (int r = 0; r < Hc / SEGT; ++r) {
            const int o = t + SEGT * r;      // output element owned
            const int h = o >> 6;            // o / HDc
            float a = acc[r] * fh[h];
            for (int ec = 0; ec < n; ++ec)
                a += sc[ec][h] * V[(size_t)rowj[ec] * Hc + o];
            acc[r] = a;
        }
    }
    __syncthreads();

#pragma unroll
    for (int r = 0; r < Hc / SEGT; ++r) {
        const int o = t + SEGT * r;
        const int h = o >> 6;
        out[(size_t)seg * Hc + o] = acc[r] / fmaxf(dh[h], 1e-9f);
    }
}

// ======================= host-side launch ================================
extern "C" void kernel_launch(void* const* d_in, const int* in_sizes, int n_in,
                              void* d_out, int out_size, void* d_ws, size_t ws_size,
                              hipStream_t stream) {
    const float* X    = (const float*)d_in[0];   // (B,N,H)
    const int*   eidx = (const int*)  d_in[1];   // (4,E): b,i,j,rel
    const float* Wq   = (const float*)d_in[2];
    const float* Wk   = (const float*)d_in[3];
    const float* Wv   = (const float*)d_in[4];
    float* out = (float*)d_out;                  // (B,N,H)

    // workspace: Q | K | V | cnt | base | cursor | elist
    float* Q = (float*)d_ws;
    float* K = Q + (size_t)MROWS * Hc;
    float* V = K + (size_t)MROWS * Hc;
    int* cnt    = (int*)(V + (size_t)MROWS * Hc);
    int* base   = cnt + NSEG;
    int* cursor = base + NSEG;
    int* elist  = cursor + NSEG;

    gat_zero<<<(NSEG + 255) / 256, 256, 0, stream>>>(cnt, NSEG);
    {
        dim3 grid(Hc / 16, MROWS / 16 / 8);      // (48, 64); 8 waves/block
        gat_qkv_gemm<<<grid, 256, 0, stream>>>(X, Wq, Wk, Wv, Q, K, V);
    }
    gat_count  <<<(Ec + 255) / 256, 256, 0, stream>>>(eidx, cnt);
    gat_scan   <<<1, 1024, 0, stream>>>(cnt, base, cursor);
    gat_scatter<<<(Ec + 255) / 256, 256, 0, stream>>>(eidx, cursor, elist);
    gat_segment_attn<<<NSEG, SEGT, 0, stream>>>(Q, K, V, eidx, base, cnt, elist, out);
}